// CoGNN_56513179681088
// MI455X (gfx1250) — compile-verified
//
#include <hip/hip_runtime.h>
#include <hip/hip_bf16.h>

#define DIM   128
#define COUT  10
#define LAYERS 3
#define TAUF  0.5f
#define EPSF  1e-5f
#define LSTR  132   // 128 + 4 padding: kills 16-way LDS bank conflicts; rows stay 16B-aligned

typedef __attribute__((ext_vector_type(2))) float v2f;
typedef __attribute__((ext_vector_type(8))) float v8f;

__device__ __forceinline__ float wave_sum(float v) {
#pragma unroll
  for (int off = 16; off >= 1; off >>= 1) v += __shfl_xor(v, off, 32);
  return v;
}

// Async DMA: global memory -> LDS, 16 bytes, no VGPR staging (ASYNCcnt path).
__device__ __forceinline__ void async_copy_b128(unsigned lds_byte_off, const float* gsrc) {
  asm volatile("global_load_async_to_lds_b128 %0, %1, off"
               :: "v"(lds_byte_off), "v"(gsrc) : "memory");
}
__device__ __forceinline__ void wait_async_zero() {
  asm volatile("s_wait_asynccnt 0" ::: "memory");
}

// ---------------------------------------------------------------- zero fill
__global__ void zero_kernel(float* __restrict__ p, long n) {
  long i = (long)blockIdx.x * blockDim.x + threadIdx.x;
  long stride = (long)gridDim.x * blockDim.x;
  for (; i < n; i += stride) p[i] = 0.0f;
}

// ---------------------------------------------------------------- LayerNorm (wave per row)
__global__ void ln_kernel(const float* __restrict__ h, const float* __restrict__ g,
                          const float* __restrict__ b, float* __restrict__ hn, int n) {
  int row  = (blockIdx.x * blockDim.x + threadIdx.x) >> 5;
  int lane = threadIdx.x & 31;
  if (row >= n) return;
  float4 v = ((const float4*)(h + (size_t)row * DIM))[lane];
  float mean = wave_sum(v.x + v.y + v.z + v.w) * (1.0f / DIM);
  float dx = v.x - mean, dy = v.y - mean, dz = v.z - mean, dw = v.w - mean;
  float var = wave_sum(dx * dx + dy * dy + dz * dz + dw * dw) * (1.0f / DIM);
  float rs = rsqrtf(var + EPSF);
  float4 gg = ((const float4*)g)[lane];
  float4 bb = ((const float4*)b)[lane];
  float4 o;
  o.x = dx * rs * gg.x + bb.x;
  o.y = dy * rs * gg.y + bb.y;
  o.z = dz * rs * gg.z + bb.z;
  o.w = dw * rs * gg.w + bb.w;
  ((float4*)(hn + (size_t)row * DIM))[lane] = o;
}

// ---------------------------------------------------------------- agg[v] += hn[u]  (wave per edge)
__global__ void scatter_agg_kernel(const float* __restrict__ hn, const int* __restrict__ eu,
                                   const int* __restrict__ ev, float* __restrict__ agg, int E) {
  int e    = (blockIdx.x * blockDim.x + threadIdx.x) >> 5;
  int lane = threadIdx.x & 31;
  if (e >= E) return;
  int u = eu[e], v = ev[e];
  float4 val = ((const float4*)(hn + (size_t)u * DIM))[lane];
  float* dst = agg + (size_t)v * DIM + lane * 4;
  atomicAdd(dst + 0, val.x);
  atomicAdd(dst + 1, val.y);
  atomicAdd(dst + 2, val.z);
  atomicAdd(dst + 3, val.w);
}

// ---------------------------------------------------------------- Gumbel straight-through, class-0 value
__device__ __forceinline__ float st_gate0(float l0, float l1, float g0, float g1) {
  float z0 = (l0 + g0) * (1.0f / TAUF);
  float z1 = (l1 + g1) * (1.0f / TAUF);
  float hard0 = (z0 >= z1) ? 1.0f : 0.0f;   // argmax picks index 0 on ties
  float m = fmaxf(z0, z1);
  float e0 = __expf(z0 - m), e1 = __expf(z1 - m);
  float s0 = e0 / (e0 + e1);
  return (hard0 - s0) + s0;                 // exact 0.0 when gate closed
}

// ---------------------------------------------------------------- gate heads (wave per node)
__global__ void gate_kernel(const float* __restrict__ hn, const float* __restrict__ agg,
                            const float* __restrict__ Wir, const float* __restrict__ Win,
                            const float* __restrict__ bai,
                            const float* __restrict__ Wor, const float* __restrict__ Won,
                            const float* __restrict__ bao,
                            const float* __restrict__ gin_l, const float* __restrict__ gout_l,
                            float* __restrict__ in0, float* __restrict__ out0, int n) {
  int node = (blockIdx.x * blockDim.x + threadIdx.x) >> 5;
  int lane = threadIdx.x & 31;
  if (node >= n) return;
  float4 hv = ((const float4*)(hn + (size_t)node * DIM))[lane];
  float4 av = ((const float4*)(agg + (size_t)node * DIM))[lane];
  const float* hp = (const float*)&hv;
  const float* ap = (const float*)&av;
  float a0 = 0, a1 = 0, b0 = 0, b1 = 0, c0 = 0, c1 = 0, d0 = 0, d1 = 0;
#pragma unroll
  for (int j = 0; j < 4; ++j) {
    int k = lane * 4 + j;
    float x = hp[j], y = ap[j];
    a0 += x * Wir[k * 2 + 0]; a1 += x * Wir[k * 2 + 1];
    b0 += y * Win[k * 2 + 0]; b1 += y * Win[k * 2 + 1];
    c0 += x * Wor[k * 2 + 0]; c1 += x * Wor[k * 2 + 1];
    d0 += y * Won[k * 2 + 0]; d1 += y * Won[k * 2 + 1];
  }
  a0 = wave_sum(a0); a1 = wave_sum(a1);
  b0 = wave_sum(b0); b1 = wave_sum(b1);
  c0 = wave_sum(c0); c1 = wave_sum(c1);
  d0 = wave_sum(d0); d1 = wave_sum(d1);
  if (lane == 0) {
    float li0 = a0 + b0 + bai[0], li1 = a1 + b1 + bai[1];
    float lo0 = c0 + d0 + bao[0], lo1 = c1 + d1 + bao[1];
    in0[node]  = st_gate0(li0, li1, gin_l[node * 2 + 0],  gin_l[node * 2 + 1]);
    out0[node] = st_gate0(lo0, lo1, gout_l[node * 2 + 0], gout_l[node * 2 + 1]);
  }
}

// ---------------------------------------------------------------- gated scatter + state flags
__global__ void scatter_gated_kernel(const float* __restrict__ hn, const int* __restrict__ eu,
                                     const int* __restrict__ ev, const float* __restrict__ in0,
                                     const float* __restrict__ out0, float* __restrict__ aggw,
                                     int* __restrict__ bflag, int* __restrict__ lflag, int E) {
  int e    = (blockIdx.x * blockDim.x + threadIdx.x) >> 5;
  int lane = threadIdx.x & 31;
  if (e >= E) return;
  int u = eu[e], v = ev[e];
  float ew = in0[v] * out0[u];
  if (ew > 0.5f && lane == 0) { bflag[u] = 1; lflag[v] = 1; }  // idempotent stores
  if (ew != 0.0f) {
    float4 val = ((const float4*)(hn + (size_t)u * DIM))[lane];
    float* dst = aggw + (size_t)v * DIM + lane * 4;
    atomicAdd(dst + 0, ew * val.x);
    atomicAdd(dst + 1, ew * val.y);
    atomicAdd(dst + 2, ew * val.z);
    atomicAdd(dst + 3, ew * val.w);
  }
}

__global__ void states_kernel(const int* __restrict__ bflag, const int* __restrict__ lflag,
                              int* __restrict__ out, int n) {
  int i = blockIdx.x * blockDim.x + threadIdx.x;
  if (i < n) out[i] = 3 - 2 * bflag[i] - lflag[i];
}

// ---------------------------------------------------------------- fused GEMM via fp32 WMMA
// out = [relu(A@W1 (+ B@W2) + bias)] (+ A skip). One wave per 16x16 tile, 8 waves/block.
// A/B stripes (16x128) DMA'd into LDS via GLOBAL_LOAD_ASYNC_TO_LDS_B128 (ASYNCcnt).
template <bool DUAL, bool SKIP>
__global__ void gemm_kernel(const float* __restrict__ A, const float* __restrict__ B,
                            const float* __restrict__ W1, const float* __restrict__ W2,
                            const float* __restrict__ bias, float* __restrict__ out) {
  __shared__ float sA[16 * LSTR];
  __shared__ float sB[(DUAL ? 16 : 1) * LSTR];
  int row0 = blockIdx.x * 16;
  int tid = threadIdx.x;

  // Cooperative async stripe fill: 512 B128 transfers per stripe, 2 per thread.
  for (int i = tid; i < 16 * (DIM / 4); i += 256) {
    int r  = i >> 5;            // 32 x B128 per 128-float row
    int c4 = (i & 31) * 4;      // starting float column (16B aligned everywhere)
    async_copy_b128((unsigned)(uintptr_t)&sA[r * LSTR + c4],
                    &A[(size_t)(row0 + r) * DIM + c4]);
    if (DUAL)
      async_copy_b128((unsigned)(uintptr_t)&sB[r * LSTR + c4],
                      &B[(size_t)(row0 + r) * DIM + c4]);
  }
  wait_async_zero();   // own wave's DMAs landed in LDS
  __syncthreads();     // all waves' stripes visible

  int lane = tid & 31;
  int wave = tid >> 5;
  int col0 = wave * 16;          // 8 waves cover all 8 column tiles
  int m  = lane & 15;            // row index for A frag / col index for B,C frag
  int kb = (lane >> 4) * 2;      // K sub-offset per half-wave (f32 16x16x4 layout)

  v8f acc = {};
#pragma unroll 4
  for (int k = 0; k < DIM; k += 4) {
    v2f a, b;
    a.x = sA[m * LSTR + k + kb];
    a.y = sA[m * LSTR + k + kb + 1];
    b.x = W1[(size_t)(k + kb) * DIM + col0 + m];
    b.y = W1[(size_t)(k + kb + 1) * DIM + col0 + m];
    acc = __builtin_amdgcn_wmma_f32_16x16x4_f32(false, a, false, b, (short)0, acc, false, false);
  }
  if (DUAL) {
#pragma unroll 4
    for (int k = 0; k < DIM; k += 4) {
      v2f a, b;
      a.x = sB[m * LSTR + k + kb];
      a.y = sB[m * LSTR + k + kb + 1];
      b.x = W2[(size_t)(k + kb) * DIM + col0 + m];
      b.y = W2[(size_t)(k + kb + 1) * DIM + col0 + m];
      acc = __builtin_amdgcn_wmma_f32_16x16x4_f32(false, a, false, b, (short)0, acc, false, false);
    }
  }

  float bcol = bias[col0 + m];
  int rbase = (lane >> 4) * 8;   // C/D layout: VGPR j -> row j (+8 for upper half-wave)
#pragma unroll
  for (int j = 0; j < 8; ++j) {
    int r = rbase + j;
    float val = fmaxf(acc[j] + bcol, 0.0f);
    if (SKIP) val += sA[r * LSTR + col0 + m];
    out[(size_t)(row0 + r) * DIM + col0 + m] = val;
  }
}

// ---------------------------------------------------------------- decoder (wave per node)
__global__ void decoder_kernel(const float* __restrict__ hn, const float* __restrict__ Wd,
                               const float* __restrict__ bd, float* __restrict__ out, int n) {
  int node = (blockIdx.x * blockDim.x + threadIdx.x) >> 5;
  int lane = threadIdx.x & 31;
  if (node >= n) return;
  float4 hv = ((const float4*)(hn + (size_t)node * DIM))[lane];
  const float* hp = (const float*)&hv;
#pragma unroll
  for (int c = 0; c < COUT; ++c) {
    float p = 0.0f;
#pragma unroll
    for (int j = 0; j < 4; ++j) p += hp[j] * Wd[(size_t)(lane * 4 + j) * COUT + c];
    p = wave_sum(p);
    if (lane == 0) out[(size_t)node * COUT + c] = p + bd[c];
  }
}

// ---------------------------------------------------------------- driver
extern "C" void kernel_launch(void* const* d_in, const int* in_sizes, int n_in,
                              void* d_out, int out_size, void* d_ws, size_t ws_size,
                              hipStream_t stream) {
  const float* x        = (const float*)d_in[0];
  const int*   ei       = (const int*)d_in[1];
  const float* W_enc    = (const float*)d_in[2];
  const float* b_enc    = (const float*)d_in[3];
  const float* ln_g     = (const float*)d_in[4];
  const float* ln_b     = (const float*)d_in[5];
  const float* Wr       = (const float*)d_in[6];
  const float* Wn       = (const float*)d_in[7];
  const float* bl       = (const float*)d_in[8];
  const float* Wa_in_r  = (const float*)d_in[9];
  const float* Wa_in_n  = (const float*)d_in[10];
  const float* ba_in    = (const float*)d_in[11];
  const float* Wa_out_r = (const float*)d_in[12];
  const float* Wa_out_n = (const float*)d_in[13];
  const float* ba_out   = (const float*)d_in[14];
  const float* W_dec    = (const float*)d_in[15];
  const float* b_dec    = (const float*)d_in[16];
  const float* gin      = (const float*)d_in[17];
  const float* gout     = (const float*)d_in[18];

  const int N = in_sizes[0] / DIM;   // 50000 (divisible by 16)
  const int E = in_sizes[1] / 2;     // 800000
  const int* eu = ei;
  const int* ev = ei + E;

  // workspace layout (fp32): h | hn | agg(/aggw) | in0 | out0 | bflag | lflag
  float* ws   = (float*)d_ws;
  float* h    = ws;
  float* hn   = h   + (size_t)N * DIM;
  float* agg  = hn  + (size_t)N * DIM;
  float* in0  = agg + (size_t)N * DIM;
  float* out0 = in0 + N;
  int*   bflag = (int*)(out0 + N);
  int*   lflag = bflag + N;

  float* result = (float*)d_out;
  int*   states = (int*)(result + (size_t)N * COUT);

  const int mtiles       = N / 16;        // GEMM row tiles
  const int node_blocks  = (N + 7) / 8;   // 8 waves (nodes) per 256-thread block
  const int edge_blocks  = (E + 7) / 8;   // 8 waves (edges) per 256-thread block

  // encoder: h = relu(x @ W_enc + b_enc)
  gemm_kernel<false, false><<<mtiles, 256, 0, stream>>>(x, nullptr, W_enc, nullptr, b_enc, h);

  for (int l = 0; l < LAYERS; ++l) {
    ln_kernel<<<node_blocks, 256, 0, stream>>>(h, ln_g, ln_b, hn, N);
    zero_kernel<<<512, 256, 0, stream>>>(agg, (long)N * DIM);
    scatter_agg_kernel<<<edge_blocks, 256, 0, stream>>>(hn, eu, ev, agg, E);
    gate_kernel<<<node_blocks, 256, 0, stream>>>(hn, agg, Wa_in_r, Wa_in_n, ba_in,
                                                 Wa_out_r, Wa_out_n, ba_out,
                                                 gin + (size_t)l * N * 2,
                                                 gout + (size_t)l * N * 2, in0, out0, N);
    zero_kernel<<<512, 256, 0, stream>>>(agg, (long)N * DIM);      // reuse as aggw
    zero_kernel<<<64, 256, 0, stream>>>((float*)bflag, 2L * N);    // bflag+lflag contiguous
    scatter_gated_kernel<<<edge_blocks, 256, 0, stream>>>(hn, eu, ev, in0, out0, agg,
                                                          bflag, lflag, E);
    states_kernel<<<(N + 255) / 256, 256, 0, stream>>>(bflag, lflag, states + (size_t)l * N, N);
    // h = hn + relu(hn @ Wr[l] + aggw @ Wn[l] + bl[l])  (fused K=256 WMMA chain)
    gemm_kernel<true, true><<<mtiles, 256, 0, stream>>>(hn, agg, Wr + (size_t)l * DIM * DIM,
                                                        Wn + (size_t)l * DIM * DIM,
                                                        bl + (size_t)l * DIM, h);
  }

  ln_kernel<<<node_blocks, 256, 0, stream>>>(h, ln_g, ln_b, hn, N);
  decoder_kernel<<<node_blocks, 256, 0, stream>>>(hn, W_dec, b_dec, result, N);
}